// InterAgg_5119601017179
// MI455X (gfx1250) — compile-verified
//
#include <hip/hip_runtime.h>
#include <math.h>
#include <stdint.h>

#define EMBED 64
#define DEG 32
#define ROW_BYTES (EMBED * 4)         // 256 B per feature row
#define WAVES_PER_BLOCK 2
#define THREADS (WAVES_PER_BLOCK * 32)

// Per-wave LDS staging: 3 relations * 32 rows * 256B = 24 KB; 48 KB per block.
__global__ __launch_bounds__(THREADS)
void interagg_async_kernel(const float* __restrict__ features,
                           const float* __restrict__ alpha,
                           const int*   __restrict__ nodes,
                           const int*   __restrict__ n1,
                           const int*   __restrict__ n2,
                           const int*   __restrict__ n3,
                           float*       __restrict__ out,
                           int nbatch)
{
    __shared__ float lds[WAVES_PER_BLOCK][3][DEG][EMBED];

    const int lane   = threadIdx.x & 31;
    const int wave   = threadIdx.x >> 5;
    const int node_i = blockIdx.x * WAVES_PER_BLOCK + wave;
    if (node_i >= nbatch) return;

    const int node = nodes[node_i];

    // One coalesced index load per relation; slots distributed across lanes.
    const int i1 = n1[node_i * DEG + lane];
    const int i2 = n2[node_i * DEG + lane];
    const int i3 = n3[node_i * DEG + lane];

    // Async gather: each instruction moves 2 rows (512B) — lane l handles
    // row (2p + l>>4), 16-byte chunk (l&15).
    const int sub    = lane >> 4;          // which row of the pair
    const int chunk  = lane & 15;          // 16B chunk within the 256B row
    const uint32_t lds_base =
        (uint32_t)(uintptr_t)(&lds[wave][0][0][0]);  // low 32 bits = LDS offset

    #pragma unroll
    for (int rel = 0; rel < 3; ++rel) {
        const int myidx = (rel == 0) ? i1 : ((rel == 1) ? i2 : i3);
        #pragma unroll
        for (int p = 0; p < DEG / 2; ++p) {
            const int row = 2 * p + sub;
            const int idx = __shfl(myidx, row, 32);
            const float* gsrc = features + (size_t)idx * EMBED + chunk * 4;
            const uint32_t ldst = lds_base +
                (uint32_t)(rel * DEG * ROW_BYTES + row * ROW_BYTES + chunk * 16);
            asm volatile("global_load_async_to_lds_b128 %0, %1, off"
                         :: "v"(ldst), "v"(gsrc) : "memory");
        }
    }

    // Overlap with async gathers: self row + per-feature softmax of alpha.
    const int d0 = 2 * lane;  // this lane's feature pair within 64
    const float2 self = *(const float2*)(features + (size_t)node * EMBED + d0);

    float wlo0[3], wlo1[3], whi0[3], whi1[3];
    {
        const float* a;
        float m, e0, e1, e2, inv;
        #define SMAX3(dst, drow)                                        \
            a  = alpha + (drow) * 3;                                    \
            m  = fmaxf(a[0], fmaxf(a[1], a[2]));                        \
            e0 = __expf(a[0] - m); e1 = __expf(a[1] - m);               \
            e2 = __expf(a[2] - m);                                      \
            inv = 1.0f / (e0 + e1 + e2);                                \
            dst[0] = e0 * inv; dst[1] = e1 * inv; dst[2] = e2 * inv;
        SMAX3(wlo0, d0)
        SMAX3(wlo1, d0 + 1)
        SMAX3(whi0, 64 + d0)
        SMAX3(whi1, 64 + d0 + 1)
        #undef SMAX3
    }

    // Drain all async gathers for this wave, then reduce from LDS.
    asm volatile("s_wait_asynccnt 0" ::: "memory");

    float aggx[3], aggy[3];
    #pragma unroll
    for (int rel = 0; rel < 3; ++rel) {
        float sx = 0.0f, sy = 0.0f;
        #pragma unroll
        for (int j = 0; j < DEG; ++j) {
            const float2 v = *(const float2*)(&lds[wave][rel][j][d0]);
            sx += v.x;
            sy += v.y;
        }
        aggx[rel] = sx * (1.0f / DEG);
        aggy[rel] = sy * (1.0f / DEG);
    }

    // aggregated[0:64]   = relu(self[d]) * sum_r w[d][r]
    // aggregated[64:128] = sum_r relu(agg_r[d]) * w[64+d][r]
    const float rsx = fmaxf(self.x, 0.0f);
    const float rsy = fmaxf(self.y, 0.0f);
    const float a1x = rsx * (wlo0[0] + wlo0[1] + wlo0[2]);
    const float a1y = rsy * (wlo1[0] + wlo1[1] + wlo1[2]);
    const float a2x = fmaxf(aggx[0], 0.0f) * whi0[0] +
                      fmaxf(aggx[1], 0.0f) * whi0[1] +
                      fmaxf(aggx[2], 0.0f) * whi0[2];
    const float a2y = fmaxf(aggy[0], 0.0f) * whi1[0] +
                      fmaxf(aggy[1], 0.0f) * whi1[1] +
                      fmaxf(aggy[2], 0.0f) * whi1[2];

    // Output row: [ self(64) | aggregated(128) ] = 192 floats.
    float* orow = out + (size_t)node_i * 192;
    *(float2*)(orow + d0)        = self;
    *(float2*)(orow + 64 + d0)   = make_float2(a1x, a1y);
    *(float2*)(orow + 128 + d0)  = make_float2(a2x, a2y);
}

extern "C" void kernel_launch(void* const* d_in, const int* in_sizes, int n_in,
                              void* d_out, int out_size, void* d_ws, size_t ws_size,
                              hipStream_t stream) {
    const float* features = (const float*)d_in[0];
    const float* alpha    = (const float*)d_in[1];
    const int*   nodes    = (const int*)d_in[2];
    const int*   n1       = (const int*)d_in[3];
    const int*   n2       = (const int*)d_in[4];
    const int*   n3       = (const int*)d_in[5];
    float*       out      = (float*)d_out;

    const int nbatch = in_sizes[2];  // 8192
    const int blocks = (nbatch + WAVES_PER_BLOCK - 1) / WAVES_PER_BLOCK;

    interagg_async_kernel<<<blocks, THREADS, 0, stream>>>(
        features, alpha, nodes, n1, n2, n3, out, nbatch);
}